// TopKTopPSampler_4801773436957
// MI455X (gfx1250) — compile-verified
//
#include <hip/hip_runtime.h>
#include <hip/hip_bf16.h>
#include <stdint.h>

// TopK+TopP+Gumbel-max sampler for MI455X (gfx1250, wave32).
// One 1024-thread block (32 waves) per row; memory-bound (~200MB mandatory
// traffic -> ~8.5us floor at 23.3 TB/s; selection re-reads ride the 192MB L2).
// CDNA5 paths: TDM tensor_load_to_lds (+ s_wait_tensorcnt) for histogram
// streaming, per-lane global_load_async_to_lds_b128 (+ s_wait_asynccnt) for
// the gather pass, global_prefetch_b8 in the final pass. Both streams are
// double-buffered to hide DMA latency behind LDS processing.

#define B_ROWS    128
#define VOCAB     128000
#define NTHREADS  1024
#define TILE_ELEMS 4096                 // 16 KB tile
#define FULL_TILES 31                   // 31*4096 = 126976
#define TAIL_BASE  126976               // remaining 1024 elems: 1 per thread
#define CAND_CAP   2048                 // top-k candidates (k <= 1024 + ties)

typedef unsigned int v4u __attribute__((ext_vector_type(4)));
typedef int          v8i __attribute__((ext_vector_type(8)));
typedef int          v4i __attribute__((ext_vector_type(4)));

// monotonic float -> uint key (larger float => larger key)
__device__ __forceinline__ uint32_t f2ord(float f) {
  uint32_t u = __float_as_uint(f);
  return (u & 0x80000000u) ? ~u : (u | 0x80000000u);
}

// ---- per-lane async DMA 16B: global -> LDS (ASYNCcnt) ----
__device__ __forceinline__ void async_tile_load(uint32_t lds_byte_off, const float* gptr) {
  asm volatile("global_load_async_to_lds_b128 %0, %1, off"
               :
               : "v"(lds_byte_off), "v"(gptr)
               : "memory");
}
__device__ __forceinline__ void wait_async0() {
  asm volatile("s_wait_asynccnt 0x0" ::: "memory");
}
__device__ __forceinline__ void wait_async1() {
  asm volatile("s_wait_asynccnt 0x1" ::: "memory");
}

// ---- TDM: one DMA moves a whole 16KB tile global -> LDS (TENSORcnt) ----
// D# group0: [1:0]=count=1, [63:32]=lds_addr, [120:64]=global_addr, [127:126]=type=2
// D# group1: data_size=2 (4B elems), tensor_dim0=tile_dim0=4096, 1-D tile.
__device__ __forceinline__ void tdm_tile_load(uint32_t lds_byte_off, const float* gptr) {
  uint64_t ga = (uint64_t)(uintptr_t)gptr;
  v4u g0;
  g0.x = 1u;                                   // count=1, no gather, user mode
  g0.y = lds_byte_off;                         // lds_addr
  g0.z = (uint32_t)ga;                         // global_addr[31:0]
  g0.w = (uint32_t)(ga >> 32) | (2u << 30);    // global_addr[56:32] | type=2
  v8i g1;
  g1[0] = (int)(2u << 16);                     // workgroup_mask=0, data_size=2 (4B)
  g1[1] = (int)((TILE_ELEMS & 0xFFFFu) << 16); // tensor_dim0[15:0]=4096
  g1[2] = (int)(1u << 16);                     // tensor_dim0[31:16]=0, tensor_dim1=1
  g1[3] = (int)((TILE_ELEMS & 0xFFFFu) << 16); // tile_dim0=4096
  g1[4] = 1;                                   // tile_dim1=1, tile_dim2=0
  g1[5] = TILE_ELEMS;                          // tensor_dim0_stride
  g1[6] = 0;
  g1[7] = 0;
  v4i z4 = {};
  v8i z8 = {};
  __builtin_amdgcn_tensor_load_to_lds(g0, g1, z4, z4, z8, 0);
}

__global__ __launch_bounds__(NTHREADS)
void topk_topp_sample_kernel(const float* __restrict__ logits,
                             const int*   __restrict__ kk,
                             const float* __restrict__ pp,
                             const float* __restrict__ q,
                             float*       __restrict__ out) {
  __shared__ __align__(16) float s_tile[2 * TILE_ELEMS]; // 32 KB double buffer
  __shared__ float    s_cand[CAND_CAP];                  // 8 KB candidate logits
  __shared__ uint32_t s_hist[256];
  __shared__ uint32_t s_r, s_prefix, s_detmask, s_cnt, s_m;
  __shared__ float    s_thr2, s_invZ, s_rowmax;

  const int tid = threadIdx.x;
  const int row = blockIdx.x;
  const float* rowlog = logits + (size_t)row * VOCAB;
  const float* rowq   = q      + (size_t)row * VOCAB;

  int kv = kk[row];
  if (kv < 1) kv = 1;
  if (kv > VOCAB) kv = VOCAB;
  const float pv = pp[row];

  if (tid == 0) { s_r = (uint32_t)kv; s_prefix = 0u; s_detmask = 0u; s_cnt = 0u; }

  const uint32_t lds_base = (uint32_t)(uintptr_t)&s_tile[0];
  const uint32_t lds_lane = lds_base + (uint32_t)tid * 16u;   // per-lane 16B slot

  __syncthreads();

  // ============ Phase 1: radix-select key of the k-th largest logit ============
  // Streaming via TDM (wave 0 drives the Tensor Data Mover, double-buffered).
  for (int pass = 0; pass < 4; ++pass) {
    const int shift = 24 - 8 * pass;
    if (tid < 256) s_hist[tid] = 0u;
    const uint32_t det = s_detmask, pref = s_prefix;
    if (tid < 32) tdm_tile_load(lds_base, rowlog);            // preload tile 0
    __syncthreads();

    for (int tile = 0; tile < FULL_TILES; ++tile) {
      if (tid < 32) {
        if (tile + 1 < FULL_TILES) {
          tdm_tile_load(lds_base + ((tile + 1) & 1) * (TILE_ELEMS * 4u),
                        rowlog + (tile + 1) * TILE_ELEMS);
          __builtin_amdgcn_s_wait_tensorcnt(1);   // tile `tile` has landed
        } else {
          __builtin_amdgcn_s_wait_tensorcnt(0);
        }
      }
      __syncthreads();
      const float* buf = s_tile + (tile & 1) * TILE_ELEMS;
#pragma unroll
      for (int j = 0; j < 4; ++j) {
        uint32_t key = f2ord(buf[tid + j * NTHREADS]);        // bank-conflict-free
        if ((key & det) == pref)
          atomicAdd(&s_hist[(key >> shift) & 0xFFu], 1u);
      }
      __syncthreads();
    }
    {   // tail: one element per thread, plain load
      uint32_t key = f2ord(rowlog[TAIL_BASE + tid]);
      if ((key & det) == pref)
        atomicAdd(&s_hist[(key >> shift) & 0xFFu], 1u);
    }
    __syncthreads();
    if (tid == 0) {   // scan 256 bins from the top to locate the rank-r digit
      uint32_t r = s_r, cum = 0u; int bsel = 0;
      for (int b = 255; b >= 0; --b) {
        uint32_t c = s_hist[b];
        if (cum + c >= r) { bsel = b; s_r = r - cum; break; }
        cum += c;
      }
      s_prefix  = pref | ((uint32_t)bsel << shift);
      s_detmask = det  | (0xFFu << shift);
    }
    __syncthreads();
  }
  const uint32_t thr_key = s_prefix;   // exact key of k-th largest => top-k thr

  // ====== Phase 2: gather candidates (logit >= thr) via async-to-LDS ======
  async_tile_load(lds_lane, rowlog + tid * 4);                // preload tile 0
  for (int tile = 0; tile < FULL_TILES; ++tile) {
    if (tile + 1 < FULL_TILES) {
      async_tile_load(lds_lane + ((tile + 1) & 1) * (TILE_ELEMS * 4u),
                      rowlog + (tile + 1) * TILE_ELEMS + tid * 4);
      wait_async1();                                          // tile `tile` done
    } else {
      wait_async0();
    }
    __syncthreads();
    const float* buf = s_tile + (tile & 1) * TILE_ELEMS;
#pragma unroll
    for (int j = 0; j < 4; ++j) {
      float v = buf[tid + j * NTHREADS];
      if (f2ord(v) >= thr_key) {
        uint32_t pos = atomicAdd(&s_cnt, 1u);
        if (pos < CAND_CAP) s_cand[pos] = v;
      }
    }
    __syncthreads();
  }
  {
    float v = rowlog[TAIL_BASE + tid];
    if (f2ord(v) >= thr_key) {
      uint32_t pos = atomicAdd(&s_cnt, 1u);
      if (pos < CAND_CAP) s_cand[pos] = v;
    }
  }
  __syncthreads();
  int n_c = (int)s_cnt;
  if (n_c > CAND_CAP) n_c = CAND_CAP;

  // ============ Phase 3: bitonic sort candidates (descending) ============
  for (int i = tid; i < CAND_CAP; i += NTHREADS)
    if (i >= n_c) s_cand[i] = -INFINITY;
  __syncthreads();
  for (int ksz = 2; ksz <= CAND_CAP; ksz <<= 1) {
    for (int j = ksz >> 1; j > 0; j >>= 1) {
      for (int t = tid; t < CAND_CAP; t += NTHREADS) {
        int partner = t ^ j;
        if (partner > t) {
          float a = s_cand[t], b = s_cand[partner];
          bool doswap = ((t & ksz) == 0) ? (a < b) : (a > b);  // descending net
          if (doswap) { s_cand[t] = b; s_cand[partner] = a; }
        }
      }
      __syncthreads();
    }
  }

  // ============ Phase 4: exp + inclusive scan -> top-p cutoff ============
  const float rowmax = s_cand[0];
  float* bufA = s_tile;
  float* bufB = s_tile + CAND_CAP;
  for (int i = tid; i < CAND_CAP; i += NTHREADS)
    bufA[i] = (i < n_c) ? __expf(s_cand[i] - rowmax) : 0.0f;
  __syncthreads();
  for (int off = 1; off < CAND_CAP; off <<= 1) {     // Hillis-Steele
    for (int i = tid; i < CAND_CAP; i += NTHREADS)
      bufB[i] = bufA[i] + ((i >= off) ? bufA[i - off] : 0.0f);
    __syncthreads();
    float* t2 = bufA; bufA = bufB; bufB = t2;
  }
  const float Ztot = bufA[n_c - 1];
  const float pZ   = pv * Ztot;
  if (tid == 0) s_m = (uint32_t)n_c;
  __syncthreads();
  // keep rank i (desc) iff mass of strictly-greater tokens < p  <=>  excl(i) < p*Ztot
  for (int i = tid + 1; i < n_c; i += NTHREADS)
    if (bufA[i - 1] >= pZ) atomicMin(&s_m, (uint32_t)i);
  __syncthreads();
  const int m = (int)s_m;                            // #kept tokens (>=1: max kept)
  if (tid == 0) {
    s_thr2   = s_cand[m - 1];                        // smallest kept logit
    s_rowmax = rowmax;
    float Zk = bufA[m - 1];                          // kept-set normalizer
    s_invZ   = (Zk > 0.0f) ? (1.0f / Zk) : 0.0f;
  }
  __syncthreads();

  // ======= Phase 5: write probs + Gumbel-max (exponential) argmax =======
  const float thr2 = s_thr2, invZ = s_invZ, rmax = s_rowmax;
  float* outp = out + B_ROWS + (size_t)row * VOCAB;
  float best = -1.0f; int besti = 0;
  for (int v = tid; v < VOCAB; v += NTHREADS) {
    if (v + NTHREADS < VOCAB) {
      __builtin_prefetch(rowlog + v + NTHREADS, 0, 0);  // global_prefetch_b8
      __builtin_prefetch(rowq   + v + NTHREADS, 0, 0);
    }
    float l  = rowlog[v];
    float pr = (l >= thr2) ? __expf(l - rmax) * invZ : 0.0f;
    outp[v] = pr;
    float sc = (pr > 0.0f) ? pr / (-__logf(rowq[v])) : 0.0f;
    if (sc > best || (sc == best && v < besti)) { best = sc; besti = v; }
  }
  __syncthreads();
  float* s_rs = s_tile;                    // reuse LDS for the final reduce
  int*   s_ri = (int*)(s_tile + NTHREADS);
  s_rs[tid] = best; s_ri[tid] = besti;
  __syncthreads();
  for (int off = NTHREADS >> 1; off > 0; off >>= 1) {
    if (tid < off) {
      float ob = s_rs[tid + off]; int oi = s_ri[tid + off];
      if (ob > s_rs[tid] || (ob == s_rs[tid] && oi < s_ri[tid])) {
        s_rs[tid] = ob; s_ri[tid] = oi;
      }
    }
    __syncthreads();
  }
  if (tid == 0) out[row] = (float)s_ri[0];   // sampled token id (as float)
}

extern "C" void kernel_launch(void* const* d_in, const int* in_sizes, int n_in,
                              void* d_out, int out_size, void* d_ws, size_t ws_size,
                              hipStream_t stream) {
  const float* logits = (const float*)d_in[0];   // [B, V] f32
  const int*   k      = (const int*)  d_in[1];   // [B] i32
  const float* p      = (const float*)d_in[2];   // [B] f32
  const float* q      = (const float*)d_in[3];   // [B, V] f32
  float* out = (float*)d_out;                    // [B] sampled ++ [B*V] probs

  topk_topp_sample_kernel<<<dim3(B_ROWS), dim3(NTHREADS), 0, stream>>>(
      logits, k, p, q, out);
}